// MambaBlock_87333864997504
// MI455X (gfx1250) — compile-verified
//
#include <hip/hip_runtime.h>
#include <hip/hip_bf16.h>
#include <math.h>

// ---------------- problem constants (from reference) ----------------
static constexpr int BATCH   = 4;
static constexpr int SEQLEN  = 2048;
static constexpr int D_MODEL = 256;
static constexpr int D_INNER = 1024;
static constexpr int D_STATE = 16;
static constexpr int D_CONV  = 4;
static constexpr int DT_RANK = 16;
static constexpr int N_LAYER = 2;
static constexpr int XDB_N   = DT_RANK + 2 * D_STATE;   // 48
static constexpr int ROWS    = BATCH * SEQLEN;          // 8192
static constexpr float EPS   = 1e-5f;

typedef float v2f __attribute__((ext_vector_type(2)));
typedef float v8f __attribute__((ext_vector_type(8)));

// =====================================================================
// WMMA fp32 GEMM:  C[M,N] = A[M,K] * W[N,K]^T   (row-major A, row-major W)
// Each wave computes a 16 x (16*NTW) tile using V_WMMA_F32_16X16X4_F32.
// A-frag / B-frag per lane are contiguous float2 along K:
//   lane<16  -> K = k, k+1   ; lane>=16 -> K = k+2, k+3
// =====================================================================
template <int NTW>
__global__ __launch_bounds__(256) void gemm_wmma_f32(
    const float* __restrict__ A, const float* __restrict__ W,
    float* __restrict__ C, int M, int N, int K)
{
    const int wave = threadIdx.x >> 5;
    const int lane = threadIdx.x & 31;
    const int nTilesN = N / (16 * NTW);
    const int waveId = blockIdx.x * 8 + wave;
    const int totalTiles = (M >> 4) * nTilesN;
    if (waveId >= totalTiles) return;            // wave-uniform exit

    const int mTile = waveId / nTilesN;
    const int nTile = waveId % nTilesN;

    const int hi = lane >> 4;                    // 0 or 1
    const int lr = lane & 15;

    const float* Ap = A + (size_t)(mTile * 16 + lr) * K + 2 * hi;
    const float* Wp[NTW];
#pragma unroll
    for (int i = 0; i < NTW; ++i)
        Wp[i] = W + (size_t)(nTile * 16 * NTW + i * 16 + lr) * K + 2 * hi;

    v8f acc[NTW] = {};

    for (int k = 0; k < K; k += 16) {
        __builtin_prefetch(Ap + k + 64, 0, 1);
#pragma unroll
        for (int kk = 0; kk < 16; kk += 4) {
            v2f afrag = *(const v2f*)(Ap + k + kk);
#pragma unroll
            for (int i = 0; i < NTW; ++i) {
                v2f bfrag = *(const v2f*)(Wp[i] + k + kk);
                acc[i] = __builtin_amdgcn_wmma_f32_16x16x4_f32(
                    false, afrag, false, bfrag, (short)0, acc[i], false, false);
            }
        }
    }

    // C/D layout: VGPR v: lanes 0-15 -> row m0+v, lanes 16-31 -> row m0+8+v; col = n0+lr
    const int m0 = mTile * 16 + hi * 8;
#pragma unroll
    for (int i = 0; i < NTW; ++i) {
        float* Cp = C + (size_t)m0 * N + nTile * 16 * NTW + i * 16 + lr;
#pragma unroll
        for (int v = 0; v < 8; ++v)
            Cp[(size_t)v * N] = acc[i][v];
    }
}

// =====================================================================
// RMSNorm over last dim (256), optional residual add. One wave per row.
// =====================================================================
__global__ __launch_bounds__(256) void rmsnorm_kernel(
    const float* __restrict__ a, const float* __restrict__ b,
    const float* __restrict__ w, float* __restrict__ out, int rows)
{
    const int wave = threadIdx.x >> 5;
    const int lane = threadIdx.x & 31;
    const int row  = blockIdx.x * 8 + wave;
    if (row >= rows) return;

    const float* ar = a + (size_t)row * D_MODEL;
    const float* br = b ? b + (size_t)row * D_MODEL : nullptr;

    float v[8];
    float ss = 0.f;
#pragma unroll
    for (int i = 0; i < 8; ++i) {
        int c = i * 32 + lane;
        float t = ar[c];
        if (br) t += br[c];
        v[i] = t;
        ss += t * t;
    }
#pragma unroll
    for (int off = 16; off >= 1; off >>= 1)
        ss += __shfl_xor(ss, off, 32);

    const float scale = rsqrtf(ss * (1.0f / D_MODEL) + EPS);
    float* orow = out + (size_t)row * D_MODEL;
#pragma unroll
    for (int i = 0; i < 8; ++i) {
        int c = i * 32 + lane;
        orow[c] = v[i] * scale * w[c];
    }
}

// =====================================================================
// Depthwise causal conv (4 taps) + bias + SiLU.  xc = xz[..., 0:1024]
// =====================================================================
__global__ __launch_bounds__(256) void conv_silu_kernel(
    const float* __restrict__ xz, const float* __restrict__ cw,
    const float* __restrict__ cb, float* __restrict__ u)
{
    const long idx = (long)blockIdx.x * 256 + threadIdx.x;
    const int d = (int)(idx % D_INNER);
    const long bl = idx / D_INNER;
    const int l = (int)(bl % SEQLEN);
    const int b = (int)(bl / SEQLEN);

    float w0 = cw[d * 4 + 0], w1 = cw[d * 4 + 1], w2 = cw[d * 4 + 2], w3 = cw[d * 4 + 3];
    const float* base = xz + ((size_t)(b * SEQLEN) + l) * (2 * D_INNER) + d;

    float s = cb[d];
    if (l >= 3) s += w0 * base[-3 * 2 * D_INNER];
    if (l >= 2) s += w1 * base[-2 * 2 * D_INNER];
    if (l >= 1) s += w2 * base[-1 * 2 * D_INNER];
    s += w3 * base[0];

    float sig = 1.0f / (1.0f + __expf(-s));
    u[idx] = s * sig;
}

// =====================================================================
// Fused: dt = softplus(xdb[:, :16] @ W_dt^T + b_dt); selective scan;
// y = scan + D*u; y *= silu(z); written IN PLACE over u.
// One thread per (b, d) channel; 16-state recurrence in registers.
// =====================================================================
__global__ __launch_bounds__(256) void scan_kernel(
    float* __restrict__ u,                 // [B, L, 1024] in/out
    const float* __restrict__ xdb,         // [B, L, 48]
    const float* __restrict__ xz,          // [B, L, 2048] (z at +1024)
    const float* __restrict__ W_dt,        // [1024, 16]
    const float* __restrict__ b_dt,        // [1024]
    const float* __restrict__ A_log,       // [1024, 16]
    const float* __restrict__ Dp)          // [1024]
{
    const int b = blockIdx.x >> 2;                       // 4 blocks per batch
    const int d = (blockIdx.x & 3) * 256 + threadIdx.x;  // channel

    float wdt[DT_RANK], Aa[D_STATE], h[D_STATE];
#pragma unroll
    for (int r = 0; r < DT_RANK; ++r) wdt[r] = W_dt[d * DT_RANK + r];
#pragma unroll
    for (int s = 0; s < D_STATE; ++s) {
        Aa[s] = -__expf(A_log[d * D_STATE + s]);
        h[s] = 0.f;
    }
    const float bdt = b_dt[d];
    const float dskip = Dp[d];

    size_t u_off = (size_t)(b * SEQLEN) * D_INNER + d;
    size_t z_off = (size_t)(b * SEQLEN) * (2 * D_INNER) + D_INNER + d;
    const float* xr = xdb + (size_t)(b * SEQLEN) * XDB_N;

    for (int t = 0; t < SEQLEN; ++t) {
        // dt = softplus(dot16(xdb_row, wdt) + bdt)   (xdb row is wave-uniform)
        float acc = bdt;
#pragma unroll
        for (int r = 0; r < DT_RANK; ++r) acc += xr[r] * wdt[r];
        float dt = (acc > 20.f) ? acc : log1pf(__expf(acc));

        float ut = u[u_off];
        float zt = xz[z_off];
        float dtu = dt * ut;

        float y = 0.f;
#pragma unroll
        for (int s = 0; s < D_STATE; ++s) {
            float Bs = xr[DT_RANK + s];
            float Cs = xr[DT_RANK + D_STATE + s];
            float dA = __expf(dt * Aa[s]);
            h[s] = dA * h[s] + dtu * Bs;
            y += h[s] * Cs;
        }
        y += dskip * ut;

        float sig = 1.0f / (1.0f + __expf(-zt));
        u[u_off] = y * (zt * sig);

        u_off += D_INNER;
        z_off += 2 * D_INNER;
        xr += XDB_N;
    }
}

// =====================================================================
// time-flip copy (D_MODEL features)
// =====================================================================
__global__ __launch_bounds__(256) void flip_kernel(
    const float* __restrict__ in, float* __restrict__ out)
{
    const long idx = (long)blockIdx.x * 256 + threadIdx.x;
    const int c = (int)(idx % D_MODEL);
    const long bl = idx / D_MODEL;
    const int l = (int)(bl % SEQLEN);
    const int b = (int)(bl / SEQLEN);
    out[idx] = in[((size_t)(b * SEQLEN) + (SEQLEN - 1 - l)) * D_MODEL + c];
}

// write one 256-wide half of the 512-wide output, optional time flip
__global__ __launch_bounds__(256) void writeout_kernel(
    const float* __restrict__ h, float* __restrict__ out, int colOff, int flip)
{
    const long idx = (long)blockIdx.x * 256 + threadIdx.x;
    const int c = (int)(idx % D_MODEL);
    const long bl = idx / D_MODEL;
    const int l = (int)(bl % SEQLEN);
    const int b = (int)(bl / SEQLEN);
    const int lsrc = flip ? (SEQLEN - 1 - l) : l;
    out[((size_t)(b * SEQLEN) + l) * (2 * D_MODEL) + colOff + c] =
        h[((size_t)(b * SEQLEN) + lsrc) * D_MODEL + c];
}

// =====================================================================
// host orchestration
// =====================================================================
extern "C" void kernel_launch(void* const* d_in, const int* in_sizes, int n_in,
                              void* d_out, int out_size, void* d_ws, size_t ws_size,
                              hipStream_t stream)
{
    const float* x      = (const float*)d_in[0];
    const float* norm_w = (const float*)d_in[1];
    const float* W_in   = (const float*)d_in[2];
    const float* conv_w = (const float*)d_in[3];
    const float* conv_b = (const float*)d_in[4];
    const float* W_x    = (const float*)d_in[5];
    const float* W_dt   = (const float*)d_in[6];
    const float* b_dt   = (const float*)d_in[7];
    const float* A_log  = (const float*)d_in[8];
    const float* D_par  = (const float*)d_in[9];
    const float* W_out  = (const float*)d_in[10];
    float* out = (float*)d_out;

    // workspace layout (floats)
    float* ws = (float*)d_ws;
    const size_t SZ_X   = (size_t)ROWS * D_MODEL;        // 2M
    const size_t SZ_XZ  = (size_t)ROWS * 2 * D_INNER;    // 16M
    const size_t SZ_U   = (size_t)ROWS * D_INNER;        // 8M
    const size_t SZ_XDB = (size_t)ROWS * XDB_N;          // 384K
    float* buf_xflip = ws;                 // 2M
    float* buf_hn    = buf_xflip + SZ_X;   // 2M
    float* buf_h     = buf_hn + SZ_X;      // 2M
    float* buf_xz    = buf_h + SZ_X;       // 16M
    float* buf_u     = buf_xz + SZ_XZ;     // 8M
    float* buf_xdb   = buf_u + SZ_U;       // 384K
    (void)buf_xdb; (void)ws_size; (void)n_in; (void)in_sizes; (void)out_size;

    const long elemsX = (long)ROWS * D_MODEL;
    const int  gridX  = (int)(elemsX / 256);
    const int  gridU  = (int)(((long)ROWS * D_INNER) / 256);

    auto run_dir = [&](const float* xin, int base, int colOff, int flipOut) {
        for (int i = 0; i < N_LAYER; ++i) {
            const int j = base + i;
            const float* na = (i == 0) ? xin : buf_h;
            const float* nb = (i == 0) ? nullptr : xin;

            rmsnorm_kernel<<<ROWS / 8, 256, 0, stream>>>(
                na, nb, norm_w + (size_t)j * D_MODEL, buf_hn, ROWS);

            // xz = hn @ W_in^T : M=8192, N=2048, K=256, NTW=4 -> 2048 blocks
            gemm_wmma_f32<4><<<(ROWS / 16) * (2 * D_INNER / 64) / 8, 256, 0, stream>>>(
                buf_hn, W_in + (size_t)j * 2 * D_INNER * D_MODEL, buf_xz,
                ROWS, 2 * D_INNER, D_MODEL);

            conv_silu_kernel<<<gridU, 256, 0, stream>>>(
                buf_xz, conv_w + (size_t)j * D_INNER * D_CONV,
                conv_b + (size_t)j * D_INNER, buf_u);

            // xdb = u @ W_x^T : M=8192, N=48, K=1024, NTW=3 -> 64 blocks
            gemm_wmma_f32<3><<<(ROWS / 16) * (XDB_N / 48) / 8, 256, 0, stream>>>(
                buf_u, W_x + (size_t)j * XDB_N * D_INNER, buf_xdb,
                ROWS, XDB_N, D_INNER);

            scan_kernel<<<BATCH * (D_INNER / 256), 256, 0, stream>>>(
                buf_u, buf_xdb, buf_xz,
                W_dt + (size_t)j * D_INNER * DT_RANK,
                b_dt + (size_t)j * D_INNER,
                A_log + (size_t)j * D_INNER * D_STATE,
                D_par + (size_t)j * D_INNER);

            // h = y @ W_out^T : M=8192, N=256, K=1024, NTW=4 -> 256 blocks
            gemm_wmma_f32<4><<<(ROWS / 16) * (D_MODEL / 64) / 8, 256, 0, stream>>>(
                buf_u, W_out + (size_t)j * D_MODEL * D_INNER, buf_h,
                ROWS, D_MODEL, D_INNER);
        }
        writeout_kernel<<<gridX, 256, 0, stream>>>(buf_h, out, colOff, flipOut);
    };

    // forward direction
    run_dir(x, 0, 0, 0);
    // backward direction: flip input, run, flip output into second half
    flip_kernel<<<gridX, 256, 0, stream>>>(x, buf_xflip);
    run_dir(buf_xflip, N_LAYER, D_MODEL, 1);
}